// UnitaryRNN_34978213658857
// MI455X (gfx1250) — compile-verified
//
#include <hip/hip_runtime.h>

// UnitaryRNN on MI455X (gfx1250):
//   out[b] = row 0 of  prod_t expm(skew(text[t,b]))   (32x32 fp32 matrices)
// Phase 1: wave-per-chunk expm + chunk product, fp32 WMMA 16x16x4.
// Phase 2: wave-per-batch reduction of chunk products, fp32 WMMA.

typedef float v2f __attribute__((ext_vector_type(2)));
typedef float v8f __attribute__((ext_vector_type(8)));

#define MATN    32
#define LDSTR   33          // padded LDS row stride (floats)
#define FEATN   496

// D = A * B (32x32 fp32 in LDS), optionally fused epilogue D = D*scl + I.
// Wave32, all lanes active (WMMA requires EXEC all-ones).
// Fragment layouts per CDNA5 ISA 7.12.2:
//  A 16x4 : lanes 0-15 rows M=0..15 hold K={k0,k0+1}; lanes 16-31 K={k0+2,k0+3}
//  B 4x16 : VGPR0 = row k0+2*half, VGPR1 = row k0+2*half+1, N = lane%16
//  C 16x16: VGPR r = row r + 8*half, N = lane%16
template <bool FUSE_SCALE_ADDI>
__device__ __forceinline__ void mm32(float* D, const float* A, const float* B,
                                     int lane, float scl) {
  const int half = lane >> 4;   // 0 or 1
  const int l16  = lane & 15;
#pragma unroll
  for (int ti = 0; ti < 2; ++ti) {
#pragma unroll
    for (int tj = 0; tj < 2; ++tj) {
      v8f acc = {};
#pragma unroll
      for (int k0 = 0; k0 < 32; k0 += 4) {
        v2f a, b;
        const float* ap = A + (ti * 16 + l16) * LDSTR + (k0 + 2 * half);
        a.x = ap[0];
        a.y = ap[1];
        const float* bp = B + (k0 + 2 * half) * LDSTR + (tj * 16 + l16);
        b.x = bp[0];
        b.y = bp[LDSTR];
        acc = __builtin_amdgcn_wmma_f32_16x16x4_f32(
            /*neg_a=*/false, a, /*neg_b=*/false, b,
            /*c_mod=*/(short)0, acc, /*reuse_a=*/false, /*reuse_b=*/false);
      }
#pragma unroll
      for (int r = 0; r < 8; ++r) {
        const int row = ti * 16 + 8 * half + r;
        const int col = tj * 16 + l16;
        float v = acc[r];
        if (FUSE_SCALE_ADDI) v = v * scl + ((row == col) ? 1.0f : 0.0f);
        D[row * LDSTR + col] = v;
      }
    }
  }
}

// ---------------- Phase 1: expm + chunk products -------------------------
__global__ __launch_bounds__(64) void unitary_phase1(
    const float* __restrict__ text, float* __restrict__ chunkP,
    int T, int Bsz, int NC, int CH) {
  __shared__ float lds[2][4][MATN * LDSTR];
  const int lane = threadIdx.x & 31;
  const int w    = threadIdx.x >> 5;
  const int gw   = blockIdx.x * 2 + w;          // global wave id = b*NC + c
  if (gw >= Bsz * NC) return;
  const int b = gw / NC;
  const int c = gw % NC;

  float* Y  = lds[w][0];   // scaled skew matrix
  float* R  = lds[w][1];   // Horner accumulator / expm result
  float* Tb = lds[w][2];   // staging / matmul temp
  float* P  = lds[w][3];   // chunk product

  // P = I (covers empty tail chunks)
  for (int i = 0; i < MATN; ++i)
    P[i * LDSTR + lane] = (i == lane) ? 1.0f : 0.0f;

  const int t0 = c * CH;
  const int t1 = (t0 + CH < T) ? (t0 + CH) : T;
  for (int t = t0; t < t1; ++t) {
    const long base = ((long)t * Bsz + b) * FEATN;

    // coalesced stage of 496 features into LDS
    for (int idx = lane; idx < FEATN; idx += 32)
      Tb[idx] = text[base + idx];

    // build skew-symmetric matrix: Y[i][j>i] = feat(i,j), Y[j][i] = -feat
    {
      const int i = lane;
      Y[i * LDSTR + i] = 0.0f;
      const int rowbase = i * (61 - i) / 2 - 1;   // f(i,j) = rowbase + j
      for (int j = i + 1; j < MATN; ++j) {
        float v = Tb[rowbase + j];
        Y[i * LDSTR + j] = v;
        Y[j * LDSTR + i] = -v;
      }
    }

    // inf-norm (max abs row sum) -> scaling exponent s
    float rs = 0.0f;
    for (int j = 0; j < MATN; ++j) rs += fabsf(Y[lane * LDSTR + j]);
    for (int off = 16; off > 0; off >>= 1)
      rs = fmaxf(rs, __shfl_xor(rs, off, 32));
    int s = 0;
    while (rs > 0.25f && s < 48) { rs *= 0.5f; ++s; }
    const float scale = ldexpf(1.0f, -s);
    for (int j = 0; j < MATN; ++j) Y[lane * LDSTR + j] *= scale;

    // degree-8 Taylor, Horner: R = I + Y/8; then R = (Y*R)/k + I, k=7..1
    // (unrolled so 1/k folds to a literal constant -- no v_div sequence)
    for (int j = 0; j < MATN; ++j)
      R[lane * LDSTR + j] =
          Y[lane * LDSTR + j] * 0.125f + ((j == lane) ? 1.0f : 0.0f);
#pragma unroll
    for (int k = 7; k >= 1; --k) {
      mm32<true>(Tb, Y, R, lane, 1.0f / (float)k);  // fused scale + I
      float* tmp = R; R = Tb; Tb = tmp;
    }

    // s squarings: E = E*E
    for (int q = 0; q < s; ++q) {
      mm32<false>(Tb, R, R, lane, 0.0f);
      float* tmp = R; R = Tb; Tb = tmp;
    }

    // accumulate chunk product
    if (t == t0) {
      // P currently identity: P = E by pointer exchange (no matmul)
      float* tmp = P; P = R; R = tmp;
    } else {
      mm32<false>(Y, P, R, lane, 0.0f);
      float* tmp = P; P = Y; Y = tmp;
    }
  }

  // coalesced store of chunk product
  float* outP = chunkP + (long)gw * (MATN * MATN);
  for (int i = 0; i < MATN; ++i)
    outP[i * MATN + lane] = P[i * LDSTR + lane];
}

// ---------------- Phase 2: per-batch reduction + row-0 output ------------
__global__ __launch_bounds__(64) void unitary_phase2(
    const float* __restrict__ chunkP, float* __restrict__ out,
    int Bsz, int NC) {
  __shared__ float lds[2][3][MATN * LDSTR];
  const int lane = threadIdx.x & 31;
  const int w    = threadIdx.x >> 5;
  const int b    = blockIdx.x * 2 + w;
  if (b >= Bsz) return;

  float* P  = lds[w][0];
  float* E  = lds[w][1];
  float* Tb = lds[w][2];

  const float* src = chunkP + (long)b * NC * (MATN * MATN);
  for (int i = 0; i < MATN; ++i)
    P[i * LDSTR + lane] = src[i * MATN + lane];

  for (int c = 1; c < NC; ++c) {
    const float* sc = src + (long)c * (MATN * MATN);
    for (int i = 0; i < MATN; ++i)
      E[i * LDSTR + lane] = sc[i * MATN + lane];
    mm32<false>(Tb, P, E, lane, 0.0f);
    float* tmp = P; P = Tb; Tb = tmp;
  }

  // h = e0 * P  ->  row 0 of P
  out[b * MATN + lane] = P[lane];
}

extern "C" void kernel_launch(void* const* d_in, const int* in_sizes, int n_in,
                              void* d_out, int out_size, void* d_ws,
                              size_t ws_size, hipStream_t stream) {
  const float* text = (const float*)d_in[0];
  const int Bsz = (n_in > 1 && in_sizes[1] > 0) ? in_sizes[1] : 128;
  const int T   = (int)((long)in_sizes[0] / ((long)Bsz * FEATN));

  // choose chunk count per batch so chunk products fit in workspace
  int NC = 32;
  while (NC > 1 &&
         (size_t)NC * (size_t)Bsz * MATN * MATN * sizeof(float) > ws_size)
    NC >>= 1;
  const int CH = (T + NC - 1) / NC;

  float* chunkP = (float*)d_ws;

  const int waves1  = Bsz * NC;
  const int blocks1 = (waves1 + 1) / 2;
  const int blocks2 = (Bsz + 1) / 2;

  unitary_phase1<<<blocks1, 64, 0, stream>>>(text, chunkP, T, Bsz, NC, CH);
  unitary_phase2<<<blocks2, 64, 0, stream>>>(chunkP, (float*)d_out, Bsz, NC);
}